// RPNCore_56650618634763
// MI455X (gfx1250) — compile-verified
//
#include <hip/hip_runtime.h>
#include <math.h>
#include <stdint.h>

#ifndef __has_builtin
#define __has_builtin(x) 0
#endif

#define A_TOTAL 262656
#define KCAND   4768
#define KPAD    8192
#define POSTN   1000
#define NMS_TH  0.7f
#define LVL_OFFSET 4096.0f
#define CHUNK   32            // = wave32: one wave resolves the greedy chain

static __device__ __constant__ int kLevelOff[5] = {0, 196608, 245760, 258048, 261120};
static __device__ __constant__ int kLevelN[5]   = {196608, 49152, 12288, 3072, 768};
static __device__ __constant__ int kLevelK[5]   = {1000, 1000, 1000, 1000, 768};
static __device__ __constant__ int kCandBase[5] = {0, 1000, 2000, 3000, 4000};

// monotonic float -> uint key (order-preserving for all finite floats)
__device__ __forceinline__ unsigned fkey(float f) {
    unsigned u = __float_as_uint(f);
    return (u & 0x80000000u) ? ~u : (u | 0x80000000u);
}

__device__ __forceinline__ float iou(float ax1, float ay1, float ax2, float ay2,
                                     float bx1, float by1, float bx2, float by2) {
    const float iw = fminf(ax2, bx2) - fmaxf(ax1, bx1);
    const float ih = fminf(ay2, by2) - fmaxf(ay1, by1);
    const float inter = fmaxf(iw, 0.0f) * fmaxf(ih, 0.0f);
    const float uni = (ax2 - ax1) * (ay2 - ay1) + (bx2 - bx1) * (by2 - by1) - inter;
    return inter / fmaxf(uni, 1e-9f);
}

// ---------------------------------------------------------------------------
// Kernel 1: per-(image,level) radix select of the k-th largest objectness key.
// 4 MSB-first 8-bit passes with an LDS histogram. Also zeroes compaction
// counters for kernel 2 (fresh every launch -> deterministic under replay).
// ---------------------------------------------------------------------------
__global__ void topk_select_kernel(const float* __restrict__ obj,
                                   unsigned* __restrict__ kth,
                                   int* __restrict__ tieneed,
                                   int* __restrict__ counters) {
    const int blk = blockIdx.x;
    const int img = blk / 5, lvl = blk % 5;
    const int off = kLevelOff[lvl], n = kLevelN[lvl], k = kLevelK[lvl];
    const float* src = obj + (size_t)img * A_TOTAL + off;

    __shared__ int hist[256];
    __shared__ unsigned s_pref;
    __shared__ int s_krem;
    if (threadIdx.x == 0) { s_pref = 0u; s_krem = k; }
    __syncthreads();

    for (int p = 0; p < 4; ++p) {
        const int shift = 24 - 8 * p;
        if (threadIdx.x < 256) hist[threadIdx.x] = 0;
        __syncthreads();
        const unsigned pref = s_pref;
        for (int i = threadIdx.x; i < n; i += blockDim.x) {
            if (((i & 127) == 0) && (i + 1024 < n))
                __builtin_prefetch(src + i + 1024, 0, 0);   // global_prefetch_b8
            unsigned u = fkey(src[i]);
            if (p == 0 || (u >> (shift + 8)) == pref)
                atomicAdd(&hist[(u >> shift) & 255], 1);
        }
        __syncthreads();
        if (threadIdx.x == 0) {
            int krem = s_krem, cum = 0, d = 255;
            for (; d >= 0; --d) {
                cum += hist[d];
                if (cum >= krem) break;
            }
            if (d < 0) d = 0;  // safety
            s_krem = krem - (cum - hist[d]);   // ties still needed within bin d
            s_pref = (pref << 8) | (unsigned)d;
        }
        __syncthreads();
    }

    if (threadIdx.x == 0) {
        kth[blk]     = s_pref;   // exact k-th largest key
        tieneed[blk] = s_krem;   // how many elements equal to kth we take
        counters[blk * 2 + 0] = 0;
        counters[blk * 2 + 1] = 0;
    }
}

// ---------------------------------------------------------------------------
// Kernel 2: compact the top-k of each (image,level) into the unsorted
// candidate arrays, fusing sigmoid, clip-to-image, and validity.
// Invalid candidates get key -1 (sorts behind all valid scores in (0,1)).
// ---------------------------------------------------------------------------
__global__ void topk_compact_kernel(const float* __restrict__ obj,
                                    const float* __restrict__ props,
                                    const int* __restrict__ imh,
                                    const int* __restrict__ imw,
                                    const unsigned* __restrict__ kth,
                                    const int* __restrict__ tieneed,
                                    int* __restrict__ counters,
                                    float* __restrict__ usc,
                                    float* __restrict__ ubox,
                                    int* __restrict__ ulvl) {
    const int blk = blockIdx.x;
    const int img = blk / 5, lvl = blk % 5;
    const int off = kLevelOff[lvl], n = kLevelN[lvl], k = kLevelK[lvl];
    const unsigned kv = kth[blk];
    const int tn = tieneed[blk];
    const int gcount = k - tn;
    const float W = (float)imw[0], H = (float)imh[0];
    const float* so = obj + (size_t)img * A_TOTAL + off;
    const float* sp = props + ((size_t)img * A_TOTAL + off) * 4;
    const int cbase = kCandBase[lvl];

    for (int i = threadIdx.x; i < n; i += blockDim.x) {
        const float o = so[i];
        const unsigned u = fkey(o);
        int pos = -1;
        if (u > kv) {
            pos = cbase + atomicAdd(&counters[blk * 2 + 0], 1);
        } else if (u == kv) {
            int t = atomicAdd(&counters[blk * 2 + 1], 1);
            if (t < tn) pos = cbase + gcount + t;
        }
        if (pos >= 0) {
            float x1 = sp[(size_t)i * 4 + 0], y1 = sp[(size_t)i * 4 + 1];
            float x2 = sp[(size_t)i * 4 + 2], y2 = sp[(size_t)i * 4 + 3];
            x1 = fminf(fmaxf(x1, 0.f), W); y1 = fminf(fmaxf(y1, 0.f), H);
            x2 = fminf(fmaxf(x2, 0.f), W); y2 = fminf(fmaxf(y2, 0.f), H);
            const float s = 1.0f / (1.0f + __expf(-o));
            const bool valid = (x2 - x1 >= 1e-3f) && (y2 - y1 >= 1e-3f) && (s >= 0.0f);
            const size_t q = (size_t)img * KCAND + (size_t)pos;
            usc[q] = valid ? s : -1.0f;
            ubox[q * 4 + 0] = x1; ubox[q * 4 + 1] = y1;
            ubox[q * 4 + 2] = x2; ubox[q * 4 + 3] = y2;
            ulvl[q] = lvl;
        }
    }
}

// ---------------------------------------------------------------------------
// Kernel 3: per-image descending bitonic sort of 8192 padded (key,idx) pairs
// fully in dynamic LDS (64 KB), then emit sorted score / offset-box / plain-box.
// ---------------------------------------------------------------------------
__global__ void sort_kernel(const float* __restrict__ usc,
                            const float* __restrict__ ubox,
                            const int* __restrict__ ulvl,
                            float* __restrict__ sc,
                            float* __restrict__ ob,
                            float* __restrict__ pb) {
    extern __shared__ unsigned char s_dyn[];
    float* keys = (float*)s_dyn;
    int*   vals = (int*)(s_dyn + (size_t)KPAD * sizeof(float));
    const int img = blockIdx.x;
    const int tid = threadIdx.x, nth = blockDim.x;
    const size_t base = (size_t)img * KCAND;

    for (int i = tid; i < KPAD; i += nth) {
        keys[i] = (i < KCAND) ? usc[base + i] : -2.0f;  // pad below invalid(-1)
        vals[i] = i;
    }
    __syncthreads();

    for (int ksz = 2; ksz <= KPAD; ksz <<= 1) {
        for (int js = ksz >> 1; js >= 1; js >>= 1) {
            for (int i = tid; i < KPAD; i += nth) {
                const int l = i ^ js;
                if (l > i) {
                    const bool up = ((i & ksz) == 0);
                    const float ki = keys[i], kl = keys[l];
                    const bool sw = up ? (ki < kl) : (ki > kl);  // descending
                    if (sw) {
                        keys[i] = kl; keys[l] = ki;
                        const int vi = vals[i]; vals[i] = vals[l]; vals[l] = vi;
                    }
                }
            }
            __syncthreads();
        }
    }

    for (int j = tid; j < KCAND; j += nth) {
        const int src = vals[j];
        sc[base + j] = keys[j];
        const size_t q = (base + (size_t)src) * 4;
        const float x1 = ubox[q + 0], y1 = ubox[q + 1];
        const float x2 = ubox[q + 2], y2 = ubox[q + 3];
        const float lo = (float)ulvl[base + src] * LVL_OFFSET;
        const size_t d = (base + (size_t)j) * 4;
        pb[d + 0] = x1;      pb[d + 1] = y1;
        pb[d + 2] = x2;      pb[d + 3] = y2;
        ob[d + 0] = x1 + lo; ob[d + 1] = y1 + lo;
        ob[d + 2] = x2 + lo; ob[d + 3] = y2 + lo;
    }
}

// ---------------------------------------------------------------------------
// Kernel 4: chunked greedy NMS per image (exact greedy semantics).
//  - Offset boxes are DMA'd into LDS by the Tensor Data Mover
//    (tensor_load_to_lds + s_wait_tensorcnt).
//  - Thread 0 collects the next 32 undecided candidates (amortized O(K)).
//  - Wave 0 (exactly wave32) resolves the greedy chain inside the chunk with
//    __ballot/__shfl lane ops -- zero block barriers for the serial part.
//  - All 1024 threads then suppress the undecided tail against the chunk's
//    kept boxes cached in LDS.  ~30x fewer __syncthreads than per-box greedy.
// ---------------------------------------------------------------------------
__global__ void nms_kernel(const float* __restrict__ sc,
                           const float* __restrict__ ob,
                           const float* __restrict__ pb,
                           float* __restrict__ out) {
    extern __shared__ unsigned char s_dyn[];
    float* obL = (float*)s_dyn;                                   // KCAND*4 floats
    unsigned char* state = (unsigned char*)(s_dyn + (size_t)KCAND * 4 * sizeof(float));
    __shared__ int   s_cidx[CHUNK];        // chunk candidate indices
    __shared__ float s_kbox[CHUNK * 4];    // kept chunk boxes (for tail pass)
    __shared__ int   s_nc, s_nk, s_kept, s_cur;

    const int img = blockIdx.x;
    const int tid = threadIdx.x, nth = blockDim.x;
    const size_t base = (size_t)img * KCAND;
    const float* obg = ob + base * 4;

#if defined(__AMDGCN__) && __has_builtin(__builtin_amdgcn_tensor_load_to_lds) && __has_builtin(__builtin_amdgcn_s_wait_tensorcnt)
    typedef unsigned int u32x4 __attribute__((ext_vector_type(4)));
    typedef int          i32x8 __attribute__((ext_vector_type(8)));
    typedef int          i32x4 __attribute__((ext_vector_type(4)));
    if (tid == 0) {
        const uint64_t ga = (uint64_t)(uintptr_t)obg;          // 57-bit global addr
        const unsigned lds_addr = (unsigned)(uintptr_t)obL;    // low 32 bits = LDS byte addr
        const unsigned ne = (unsigned)(KCAND * 4);             // 19072 4B elements
        // D# group 0: count=1 | lds_addr | global_addr[56:0] | type=2
        u32x4 g0 = { 1u,
                     lds_addr,
                     (unsigned)(ga & 0xFFFFFFFFu),
                     (unsigned)((ga >> 32) & 0x01FFFFFFu) | (2u << 30) };
        // D# group 1: data_size=4B; tensor_dim0=ne; tensor_dim1=1; tile_dim0=ne;
        //             tile_dim1=1; tensor_dim0_stride=ne
        i32x8 g1 = { (int)(2u << 16),
                     (int)((ne & 0xFFFFu) << 16),
                     (int)(((ne >> 16) & 0xFFFFu) | (1u << 16)),
                     (int)((ne & 0xFFFFu) << 16),
                     1,
                     (int)ne,
                     0, 0 };
        i32x4 z4 = { 0, 0, 0, 0 };
        i32x8 z8 = { 0, 0, 0, 0, 0, 0, 0, 0 };
        // amdgpu-toolchain (clang-23) 6-arg form: (g0, g1, g2, g3, g4, cpol)
        __builtin_amdgcn_tensor_load_to_lds(g0, g1, z4, z4, z8, 0);
        __builtin_amdgcn_s_wait_tensorcnt(0);
    }
#else
    for (int i = tid; i < KCAND * 4; i += nth) obL[i] = obg[i];
#endif

    for (int i = tid; i < KCAND; i += nth)
        state[i] = (sc[base + i] < 0.0f) ? (unsigned char)1 : (unsigned char)0;
    if (tid == 0) { s_kept = 0; s_cur = 0; }
    __syncthreads();

    for (;;) {
        // --- phase 1: collect next CHUNK undecided candidates (thread 0) ---
        if (tid == 0) {
            int c = s_cur, nc = 0;
            if (s_kept < POSTN) {
                while (c < KCAND && nc < CHUNK) {
                    if (state[c] == 0) s_cidx[nc++] = c;
                    c++;
                }
            }
            s_cur = c;
            s_nc = nc;
        }
        __syncthreads();
        const int nc = s_nc;
        if (nc == 0) break;                        // done or POST reached

        // --- phase 2: wave 0 resolves greedy chain within the chunk ---
        if (tid < CHUNK) {
            const int l = tid;
            const bool has = (l < nc);
            const int idx = has ? s_cidx[l] : s_cidx[0];
            const float bx1 = obL[idx * 4 + 0], by1 = obL[idx * 4 + 1];
            const float bx2 = obL[idx * 4 + 2], by2 = obL[idx * 4 + 3];
            bool alive = has;
            for (int s = 0; s < CHUNK; ++s) {
                const unsigned long long am = __ballot(alive);
                const float sx1 = __shfl(bx1, s), sy1 = __shfl(by1, s);
                const float sx2 = __shfl(bx2, s), sy2 = __shfl(by2, s);
                if (((am >> s) & 1ull) && (l > s) && alive) {
                    if (iou(bx1, by1, bx2, by2, sx1, sy1, sx2, sy2) > NMS_TH)
                        alive = false;
                }
            }
            const unsigned long long km = __ballot(alive);
            const int rank = (int)__popcll(km & ((1ull << l) - 1ull));
            const int kept0 = s_kept;              // read before lane-0 update
            if (has) state[idx] = alive ? (unsigned char)2 : (unsigned char)1;
            if (alive) {
                s_kbox[rank * 4 + 0] = bx1; s_kbox[rank * 4 + 1] = by1;
                s_kbox[rank * 4 + 2] = bx2; s_kbox[rank * 4 + 3] = by2;
                const int row = kept0 + rank;
                if (row < POSTN) {
                    float* o = out + ((size_t)img * POSTN + (size_t)row) * 5;
                    const size_t q = (base + (size_t)idx) * 4;
                    o[0] = pb[q + 0]; o[1] = pb[q + 1];
                    o[2] = pb[q + 2]; o[3] = pb[q + 3];
                    o[4] = sc[base + idx];
                }
            }
            if (l == 0) {
                s_nk = (int)__popcll(km);
                s_kept = kept0 + (int)__popcll(km);
            }
        }
        __syncthreads();

        // --- phase 3: all threads suppress the undecided tail vs kept boxes ---
        const int nk = s_nk;
        const int jstart = s_cur;                  // all chunk indices < s_cur
        if (nk > 0) {
            for (int j = jstart + tid; j < KCAND; j += nth) {
                if (state[j] == 0) {
                    const float x1 = obL[j * 4 + 0], y1 = obL[j * 4 + 1];
                    const float x2 = obL[j * 4 + 2], y2 = obL[j * 4 + 3];
                    for (int r = 0; r < nk; ++r) {
                        if (iou(x1, y1, x2, y2,
                                s_kbox[r * 4 + 0], s_kbox[r * 4 + 1],
                                s_kbox[r * 4 + 2], s_kbox[r * 4 + 3]) > NMS_TH) {
                            state[j] = 1;
                            break;
                        }
                    }
                }
            }
        }
        __syncthreads();
    }

    // zero-fill unused output rows (d_out fully written -> replay-safe)
    const int kept = (s_kept < POSTN) ? s_kept : POSTN;
    const size_t ro = ((size_t)img * POSTN + (size_t)kept) * 5;
    const int rem = (POSTN - kept) * 5;
    for (int t = tid; t < rem; t += nth) out[ro + t] = 0.0f;
}

// ---------------------------------------------------------------------------
extern "C" void kernel_launch(void* const* d_in, const int* in_sizes, int n_in,
                              void* d_out, int out_size, void* d_ws, size_t ws_size,
                              hipStream_t stream) {
    (void)n_in; (void)out_size; (void)ws_size;
    const float* obj   = (const float*)d_in[0];
    const float* props = (const float*)d_in[1];
    const int*   imh   = (const int*)d_in[2];
    const int*   imw   = (const int*)d_in[3];
    const int B = in_sizes[0] / A_TOTAL;

    // carve workspace (256B-aligned chunks)
    char* p = (char*)d_ws;
    auto carve = [&p](size_t bytes) {
        char* r = p;
        p += (bytes + 255) & ~(size_t)255;
        return r;
    };
    unsigned* kth     = (unsigned*)carve((size_t)B * 5 * sizeof(unsigned));
    int*      tieneed = (int*)carve((size_t)B * 5 * sizeof(int));
    int*      cnts    = (int*)carve((size_t)B * 5 * 2 * sizeof(int));
    float*    usc     = (float*)carve((size_t)B * KCAND * sizeof(float));
    float*    ubox    = (float*)carve((size_t)B * KCAND * 4 * sizeof(float));
    int*      ulvl    = (int*)carve((size_t)B * KCAND * sizeof(int));
    float*    sc      = (float*)carve((size_t)B * KCAND * sizeof(float));
    float*    ob      = (float*)carve((size_t)B * KCAND * 4 * sizeof(float));
    float*    pb      = (float*)carve((size_t)B * KCAND * 4 * sizeof(float));

    topk_select_kernel<<<B * 5, 1024, 0, stream>>>(obj, kth, tieneed, cnts);
    topk_compact_kernel<<<B * 5, 1024, 0, stream>>>(obj, props, imh, imw,
                                                    kth, tieneed, cnts,
                                                    usc, ubox, ulvl);
    const size_t shSort = (size_t)KPAD * (sizeof(float) + sizeof(int));   // 64 KB
    sort_kernel<<<B, 1024, shSort, stream>>>(usc, ubox, ulvl, sc, ob, pb);
    const size_t shNms = (size_t)KCAND * 4 * sizeof(float) + KCAND + 256; // ~81 KB
    nms_kernel<<<B, 1024, shNms, stream>>>(sc, ob, pb, (float*)d_out);
}